// AnisotropicCollisions_27642409517353
// MI455X (gfx1250) — compile-verified
//
#include <hip/hip_runtime.h>
#include <stdint.h>

// Problem constants (from reference)
#define NVg      512
#define NMODES   9
#define NSYS     (NMODES * 64 * 64)      // 36864 independent tridiagonal systems
#define DVc      0.015625f
#define INV_DV_SQ 4096.0f                // 1/DV^2
#define HALF_INV_DV 32.0f                // 1/(2*DV)
#define FOUR_PI_F  12.566370614359172f
#define EIGHT_PI_F 25.132741228718345f
#define Y_DT_F   1e-12f
#define THIRD    0.3333333333333333f

#define BLK   256
#define TK    32                          // tile width in k
#define NT    (NVg / TK)                  // 16 tiles
#define PITCH 36                          // 32 + 4: keeps [r][4q] 16B-aligned, odd-ish mod 64

typedef float v2f __attribute__((ext_vector_type(2)));
typedef float v8f __attribute__((ext_vector_type(8)));

// ---- CDNA5 async global->LDS (ASYNCcnt path), CDNA5 ISA ----
// Generic pointers to __shared__ carry the wave-relative LDS offset in their
// low 32 bits (ISA 10.2 aperture mapping), which is exactly what the async
// op's VDST VGPR must hold (it is added to the LDS base and the instruction offset).
__device__ __forceinline__ void ld16_to_lds(float* lds_dst, const float* gsrc) {
    unsigned lds_off = (unsigned)(uintptr_t)lds_dst;
    unsigned long long ga = (unsigned long long)(uintptr_t)gsrc;
    asm volatile("global_load_async_to_lds_b128 %0, %1, off"
                 :: "v"(lds_off), "v"(ga) : "memory");
}
// Partial wait: async loads complete in order, so ASYNCcnt <= n means the
// oldest (outstanding - n) instructions have landed.
#define WAIT_ASYNC_LE(n) asm volatile("s_wait_asynccnt %0" :: "i"(n) : "memory")

// ------------------------------------------------------------------
// Kernel 1: totals[s] = sum_k y[s,k] * v[k]   via FP32 WMMA 16x16x4.
// One wave accumulates 16 systems: A = (f*v) chunk (16x4 fp32),
// B = all-ones (4x16) -> D[:,n] = row sums for every n.
// A layout (ISA 7.12.2): lane L holds M=L%16, VGPR j holds K=2*(L/16)+j.
// ------------------------------------------------------------------
__global__ __launch_bounds__(BLK) void totals_kernel(const float* __restrict__ y,
                                                     float* __restrict__ totals) {
    const int tid  = threadIdx.x;
    const int lane = tid & 31;
    const int wave = (blockIdx.x * (BLK / 32)) + (tid >> 5);
    const int s0   = wave * 16;
    const int m    = lane & 15;
    const int half = lane >> 4;
    const float* row = y + (size_t)(s0 + m) * NVg;

#if __has_builtin(__builtin_amdgcn_wmma_f32_16x16x4_f32)
    v8f C = {0.f, 0.f, 0.f, 0.f, 0.f, 0.f, 0.f, 0.f};
    v2f Bones; Bones.x = 1.0f; Bones.y = 1.0f;
    for (int kb = 0; kb < NVg; kb += 4) {
        const int k = kb + 2 * half;
        v2f A;
        A.x = row[k]     * ((float)(k + 1) * DVc);
        A.y = row[k + 1] * ((float)(k + 2) * DVc);
        C = __builtin_amdgcn_wmma_f32_16x16x4_f32(
                false, A, false, Bones, (short)0, C, false, false);
    }
    // D(m, n=0): m<8 -> VGPR m @ lane 0 ; m>=8 -> VGPR m-8 @ lane 16
    if (lane == 0) {
#pragma unroll
        for (int r = 0; r < 8; ++r) totals[s0 + r] = C[r];
    }
    if (lane == 16) {
#pragma unroll
        for (int r = 0; r < 8; ++r) totals[s0 + 8 + r] = C[r];
    }
#else
    float acc = 0.f;
    for (int kb = 0; kb < NVg; kb += 4) {
        const int k = kb + 2 * half;
        acc += row[k]     * ((float)(k + 1) * DVc);
        acc += row[k + 1] * ((float)(k + 2) * DVc);
    }
    acc += __shfl_xor(acc, 16, 32);
    if (lane < 16) totals[s0 + m] = acc;
#endif
}

// Cooperative tile prefetch: 256 threads x 8 iters of b128 = 256 rows x 32 cols.
// 8 consecutive lanes cover one row's 128B run -> fully coalesced.
__device__ __forceinline__ void issue_tile(float (*dst)[PITCH], const float* __restrict__ g,
                                           int sysbase, int k0, int tid) {
#pragma unroll
    for (int e = 0; e < 8; ++e) {
        const int idx4 = e * BLK + tid;
        const int r = idx4 >> 3, c4 = (idx4 & 7) * 4;
        ld16_to_lds(&dst[r][c4], g + (size_t)(sysbase + r) * NVg + k0 + c4);
    }
}

// ------------------------------------------------------------------
// Kernel 2: fused weighted-cumsums + coefficients + Thomas forward.
// One lane owns one system; double-buffered async pipeline on the f tiles.
// cp -> d_out, dp -> d_ws (coalesced b128 stores via LDS staging).
// ------------------------------------------------------------------
__global__ __launch_bounds__(BLK) void fwd_kernel(const float* __restrict__ y,
                                                  const int*   __restrict__ il_arr,
                                                  const float* __restrict__ totals,
                                                  float* __restrict__ cp_out,
                                                  float* __restrict__ dp_out) {
    __shared__ float tF[2][BLK][PITCH];   // f tile (double buffered), reused for cp
    __shared__ float tDp[BLK][PITCH];     // dp tile

    const int tid     = threadIdx.x;
    const int sysbase = blockIdx.x * BLK;
    const int sys     = sysbase + tid;

    const float il  = (float)il_arr[sys >> 12];       // 4096 systems per mode
    const float lam = 0.5f * il * (il + 1.0f);
    const float T1  = totals[sys];

    float c4s = 0.f, c1s = 0.f, c2s = 0.f;  // running cumsums f*v^4, f*v, f*v^2
    float cpp = 0.f, dpp = 0.f;             // Thomas carries (0-init folds the k==0 case)

    issue_tile(tF[0], y, sysbase, 0, tid);  // prologue: tile 0 -> buf 0

    for (int t = 0; t < NT; ++t) {
        const int buf = t & 1;
        const int k0  = t * TK;
        if (t + 1 < NT) {
            issue_tile(tF[buf ^ 1], y, sysbase, k0 + TK, tid);
            WAIT_ASYNC_LE(8);   // current tile (oldest 8) landed; prefetch in flight
        } else {
            WAIT_ASYNC_LE(0);
        }
        __syncthreads();

#pragma unroll
        for (int kk = 0; kk < TK; ++kk) {
            const int   k     = k0 + kk;
            const float vk    = (float)(k + 1) * DVc;
            const float inv_v = 1.0f / vk;
            const float vk2   = vk * vk;

            const float f = tF[buf][tid][kk];
            c4s += f * (vk2 * vk2);
            c1s += f * vk;
            c2s += f * vk2;

            const float w   = FOUR_PI_F * inv_v;
            const float i2  = w * c4s;
            const float jm1 = w * (T1 - c1s);
            const float i0  = w * c2s;

            const float coeff1 = (2.0f * jm1 + 3.0f * i0 - i2) * THIRD * (inv_v * inv_v);
            const float coeff2 = (i2 + jm1) * THIRD * inv_v;

            const float A1 = coeff1 * HALF_INV_DV;   // coeff1/(2*DV)
            const float B2 = coeff2 * INV_DV_SQ;     // coeff2/DV^2

            const float a = (B2 - A1) * Y_DT_F;
            const float b = (EIGHT_PI_F * f + 0.5f * B2 - lam * coeff1 * inv_v) * Y_DT_F + 1.0f;
            const float c = (B2 + A1) * Y_DT_F;

            const float denom = b - a * cpp;         // cpp==0 at k==0 -> cp0=c/b, dp0=d/b
            const float rden  = 1.0f / denom;
            const float cp    = c * rden;
            const float dp    = (f - a * dpp) * rden;
            cpp = cp; dpp = dp;

            tF[buf][tid][kk] = cp;
            tDp[tid][kk]     = dp;
        }
        __syncthreads();

        // coalesced b128 stores: 8 lanes per 128B row segment
#pragma unroll
        for (int e = 0; e < 8; ++e) {
            const int idx4 = e * BLK + tid;
            const int r = idx4 >> 3, c4 = (idx4 & 7) * 4;
            const size_t go = (size_t)(sysbase + r) * NVg + k0 + c4;
            *(float4*)(cp_out + go) = *(const float4*)&tF[buf][r][c4];
            *(float4*)(dp_out + go) = *(const float4*)&tDp[r][c4];
        }
        __syncthreads();
    }
}

// ------------------------------------------------------------------
// Kernel 3: Thomas backward. x[k] = dp[k] - cp[k]*x[k+1], x init 0
// (cp[511]*0 == 0 makes k==511 uniform). cp lives in d_out and is
// overwritten in place with x, tiles walked from high k to low k with
// a double-buffered async pipeline on both cp and dp tiles.
// ------------------------------------------------------------------
__global__ __launch_bounds__(BLK) void bwd_kernel(const float* __restrict__ dp_in,
                                                  float* __restrict__ out) {
    __shared__ float tC[2][BLK][PITCH];   // cp tiles
    __shared__ float tD[2][BLK][PITCH];   // dp tiles, overwritten with x

    const int tid     = threadIdx.x;
    const int sysbase = blockIdx.x * BLK;
    float x = 0.0f;

    issue_tile(tC[0], out,   sysbase, (NT - 1) * TK, tid);
    issue_tile(tD[0], dp_in, sysbase, (NT - 1) * TK, tid);

    for (int t = NT - 1; t >= 0; --t) {
        const int buf = (NT - 1 - t) & 1;
        const int k0  = t * TK;
        if (t > 0) {
            issue_tile(tC[buf ^ 1], out,   sysbase, k0 - TK, tid);
            issue_tile(tD[buf ^ 1], dp_in, sysbase, k0 - TK, tid);
            WAIT_ASYNC_LE(16);  // current tile's 16 (8 cp + 8 dp) landed
        } else {
            WAIT_ASYNC_LE(0);
        }
        __syncthreads();

#pragma unroll
        for (int kk = TK - 1; kk >= 0; --kk) {
            x = tD[buf][tid][kk] - tC[buf][tid][kk] * x;
            tD[buf][tid][kk] = x;
        }
        __syncthreads();

#pragma unroll
        for (int e = 0; e < 8; ++e) {
            const int idx4 = e * BLK + tid;
            const int r = idx4 >> 3, c4 = (idx4 & 7) * 4;
            *(float4*)(out + (size_t)(sysbase + r) * NVg + k0 + c4) =
                *(const float4*)&tD[buf][r][c4];
        }
        __syncthreads();
    }
}

extern "C" void kernel_launch(void* const* d_in, const int* in_sizes, int n_in,
                              void* d_out, int out_size, void* d_ws, size_t ws_size,
                              hipStream_t stream) {
    (void)in_sizes; (void)n_in; (void)out_size; (void)ws_size;
    const float* y  = (const float*)d_in[0];   // (9,64,64,512) f32
    const int*   il = (const int*)d_in[1];     // (9,) i32
    float* out = (float*)d_out;                // (9,64,64,512) f32

    float* totals = (float*)d_ws;              // NSYS floats
    float* dp     = totals + NSYS;             // NSYS*NVg floats (~75 MB)

    totals_kernel<<<NSYS / 16 / (BLK / 32), BLK, 0, stream>>>(y, totals);
    fwd_kernel<<<NSYS / BLK, BLK, 0, stream>>>(y, il, totals, out, dp);
    bwd_kernel<<<NSYS / BLK, BLK, 0, stream>>>(dp, out);
}